// AGCN_item_23244363006255
// MI455X (gfx1250) — compile-verified
//
#include <hip/hip_runtime.h>
#include <hip/hip_bf16.h>

#define N_USERS 100000
#define N_ITEMS 50000
#define N_NODES 150000
#define D_LAT   64
#define D_ATTR  32
#define DIM     96
#define RAW     512

typedef __attribute__((ext_vector_type(2))) float v2f;
typedef __attribute__((ext_vector_type(8))) float v8f;

// ---------------- float4 grid-stride copy ----------------
__global__ void k_copy_f4(float4* __restrict__ dst, const float4* __restrict__ src, int n4) {
    int i = blockIdx.x * blockDim.x + threadIdx.x;
    int stride = gridDim.x * blockDim.x;
    for (; i < n4; i += stride) dst[i] = src[i];
}

// ---------------- item latent part: emb[N_USERS+i][0:64] = item_emb[i][:] ----------------
__global__ void k_copy_items(float* __restrict__ emb, const float* __restrict__ item_emb) {
    int t = blockIdx.x * blockDim.x + threadIdx.x;     // one float4 per thread
    if (t >= N_ITEMS * (D_LAT / 4)) return;
    int i  = t >> 4;            // item
    int f4 = t & 15;            // float4 within the 64-float latent part
    float4 v = ((const float4*)item_emb)[t];
    *(float4*)(emb + (size_t)(N_USERS + i) * DIM + f4 * 4) = v;
}

// ---------------- WMMA f32 GEMM: attr = missing_attr @ W_trans^T ----------------
// attr[m][n] = sum_k A[m][k] * Wt[n][k], written into emb[N_USERS+m][64+n].
// One wave computes a 16x16 tile; K=512 -> 128 x v_wmma_f32_16x16x4_f32.
__global__ void k_gemm_attr(float* __restrict__ emb,
                            const float* __restrict__ A,    // [N_ITEMS][RAW]
                            const float* __restrict__ Wt)   // [D_ATTR][RAW]
{
    const int lane = threadIdx.x & 31;
    const int w    = blockIdx.x * (blockDim.x >> 5) + (threadIdx.x >> 5);
    const int NW   = (N_ITEMS / 16) * (D_ATTR / 16);        // 6250 tiles
    if (w >= NW) return;
    const int m0 = (w >> 1) * 16;
    const int n0 = (w & 1) * 16;
    const int lm = lane & 15;   // M (for A) / N (for B)
    const int lh = lane >> 4;   // K half-select

    // 32-bit A 16x4 layout: VGPR0 = {K0 | K2}, VGPR1 = {K1 | K3}; M = lane&15.
    const float* ap = A  + (size_t)(m0 + lm) * RAW + 2 * lh;
    // B (4x16) mirrors: row K striped across lanes; N = lane&15.
    const float* bp = Wt + (size_t)(n0 + lm) * RAW + 2 * lh;

    v8f c = {};
#pragma unroll 4
    for (int k = 0; k < RAW; k += 4) {
        v2f a = *(const v2f*)(ap + k);
        v2f b = *(const v2f*)(bp + k);
        // (neg_a, A, neg_b, B, c_mod, C, reuse_a, reuse_b)
        c = __builtin_amdgcn_wmma_f32_16x16x4_f32(false, a, false, b,
                                                  (short)0, c, false, false);
    }

    // C layout: VGPR r -> M = r + 8*lh, N = lane&15.
    float* drow = emb + (size_t)(N_USERS + m0 + 8 * lh) * DIM + D_LAT + n0 + lm;
#pragma unroll
    for (int r = 0; r < 8; ++r) drow[(size_t)r * DIM] = c[r];
}

// ---------------- SpMM scatter: dst[rows[e]] += ev[e] * src[cols[e]] ----------------
// One wave per edge; 96 features = 3 per lane. src/dst are L2-resident (57.6 MB < 192 MB).
__global__ void k_spmm(float* __restrict__ dst, const float* __restrict__ src,
                       const float* __restrict__ ev, const int* __restrict__ rows,
                       const int* __restrict__ cols, int nE) {
    int e = (blockIdx.x * blockDim.x + threadIdx.x) >> 5;
    if (e >= nE) return;
    int lane = threadIdx.x & 31;
    int r = rows[e];
    int c = cols[e];
    float v = ev[e];
    const float* s = src + (size_t)c * DIM + lane * 3;
    float*       d = dst + (size_t)r * DIM + lane * 3;
    float x0 = s[0], x1 = s[1], x2 = s[2];
    atomicAdd(d + 0, v * x0);
    atomicAdd(d + 1, v * x1);
    atomicAdd(d + 2, v * x2);
}

extern "C" void kernel_launch(void* const* d_in, const int* in_sizes, int n_in,
                              void* d_out, int out_size, void* d_ws, size_t ws_size,
                              hipStream_t stream) {
    const float* missing_attr = (const float*)d_in[0];
    const float* W_trans      = (const float*)d_in[1];
    const float* user_emb     = (const float*)d_in[2];
    const float* item_emb     = (const float*)d_in[3];
    const float* edge_vals    = (const float*)d_in[4];
    const int*   rows         = (const int*)d_in[5];
    const int*   cols         = (const int*)d_in[6];
    const int nE = in_sizes[4];

    float* W = (float*)d_ws;    // ping buffer: emb0, emb2
    float* O = (float*)d_out;   // pong buffer: emb1, emb3 (final result)

    // ---- build emb0 in W ----
    {
        int n4 = N_USERS * DIM / 4;                      // contiguous user rows
        k_copy_f4<<<2048, 256, 0, stream>>>((float4*)W, (const float4*)user_emb, n4);
    }
    k_copy_items<<<(N_ITEMS * (D_LAT / 4) + 255) / 256, 256, 0, stream>>>(W, item_emb);
    {
        int nWaves = (N_ITEMS / 16) * (D_ATTR / 16);     // 6250
        int blocks = (nWaves + 3) / 4;                   // 4 waves / block
        k_gemm_attr<<<blocks, 128, 0, stream>>>(W, missing_attr, W_trans);
    }

    const int n4emb = N_NODES * DIM / 4;
    const int spmmBlocks = (int)(((long long)nE * 32 + 255) / 256);

    // ---- layer 1: W -> O ----
    k_copy_f4<<<2048, 256, 0, stream>>>((float4*)O, (const float4*)W, n4emb);
    k_spmm<<<spmmBlocks, 256, 0, stream>>>(O, W, edge_vals, rows, cols, nE);
    // ---- layer 2: O -> W ----
    k_copy_f4<<<2048, 256, 0, stream>>>((float4*)W, (const float4*)O, n4emb);
    k_spmm<<<spmmBlocks, 256, 0, stream>>>(W, O, edge_vals, rows, cols, nE);
    // ---- layer 3: W -> O (final lands in d_out) ----
    k_copy_f4<<<2048, 256, 0, stream>>>((float4*)O, (const float4*)W, n4emb);
    k_spmm<<<spmmBlocks, 256, 0, stream>>>(O, W, edge_vals, rows, cols, nE);
}